// FinalModel_3985729651483
// MI455X (gfx1250) — compile-verified
//
#include <hip/hip_runtime.h>
#include <hip/hip_bf16.h>

#define SPATIAL 9
#define BANDS   200
#define HIDDEN  1024
#define OUT_DIM 512
#define BATCH   32
#define DIN     (BANDS * SPATIAL * SPATIAL)   // 16200
#define LN_EPS  1e-5f

#define PROJ_SPLIT  10                         // 10 * 1620 = 16200, 1620 % 4 == 0
#define PROJ_KCHUNK (DIN / PROJ_SPLIT)         // 1620
#define CONV_SPLIT  8                          // 8 * 128 = 1024
#define CONV_KCHUNK (HIDDEN / CONV_SPLIT)      // 128

typedef __attribute__((ext_vector_type(2))) float v2f;
typedef __attribute__((ext_vector_type(8))) float v8f;

// Native CDNA5 transcendental tanh (VOP1 on gfx1250). Trailing v_nop covers the
// TRANS-op "1 independent instruction before result use" hazard across the
// opaque asm boundary.
__device__ __forceinline__ float fast_tanh(float x) {
    float r;
    asm volatile("v_tanh_f32 %0, %1\n\tv_nop" : "=v"(r) : "v"(x));
    return r;
}

// ---------------------------------------------------------------------------
// Wave-level 32xK @ Kx16 fp32 GEMM slab on V_WMMA_F32_16X16X4_F32.
// K is a COMPILE-TIME constant so the K-loop unrolls by 4 / software-
// pipelines (deep load clauses + partial s_wait_loadcnt before each WMMA)
// with uniform scalar control flow (EXEC stays all-ones, as WMMA requires).
// A (activations): [32 x *] row-major, base already offset to the K chunk.
// B element (k, n) = W[(n)*ldn + k*ldk + off]  (off carries K-chunk offset and
// conv-tap offset).  ISA 16x4 fp32 A layout: lanes 0-15 hold {K,K+1} for
// M=lane, lanes 16-31 hold {K+2,K+3}; B is the transposed analogue.
// ---------------------------------------------------------------------------
template <int K>
__device__ __forceinline__ void gemm32_tile(
    const float* __restrict__ Act, int lda,
    const float* __restrict__ W, long ldn, long ldk, long off,
    int n0, v8f& c0, v8f& c1)
{
    const int lane = threadIdx.x & 31;
    const int half = lane >> 4;     // 0: K,K+1   1: K+2,K+3
    const int lid  = lane & 15;

    const float* a0row = Act + (size_t)lid * lda;          // M = 0..15 tile
    const float* a1row = Act + (size_t)(lid + 16) * lda;   // M = 16..31 tile
    const float* wrow  = W + (size_t)(n0 + lid) * ldn + off;

#pragma unroll 4
    for (int k = 0; k < K; k += 4) {
        const int ka = k + 2 * half;
        v2f a0, a1, b;
        a0.x = a0row[ka];
        a0.y = a0row[ka + 1];
        a1.x = a1row[ka];
        a1.y = a1row[ka + 1];
        b.x  = wrow[(size_t)ka * ldk];
        b.y  = wrow[(size_t)(ka + 1) * ldk];
        __builtin_prefetch(wrow + (size_t)(ka + 32) * ldk, 0, 1);
        // 8 args: (neg_a, A, neg_b, B, c_mod, C, reuse_a, reuse_b)
        c0 = __builtin_amdgcn_wmma_f32_16x16x4_f32(false, a0, false, b,
                                                   (short)0, c0, false, false);
        c1 = __builtin_amdgcn_wmma_f32_16x16x4_f32(false, a1, false, b,
                                                   (short)0, c1, false, false);
    }
}

// C/D layout: VGPR r -> (M = r + 8*half, N = lid) for tile 0; +16 rows tile 1.
__device__ __forceinline__ void store_tile(float* __restrict__ out, int ldo, int n0,
                                           const v8f& c0, const v8f& c1)
{
    const int lane = threadIdx.x & 31;
    const int half = lane >> 4;
    const int lid  = lane & 15;
#pragma unroll
    for (int r = 0; r < 8; ++r) {
        out[(size_t)(r + 8 * half) * ldo + n0 + lid]      = c0[r];
        out[(size_t)(16 + r + 8 * half) * ldo + n0 + lid] = c1[r];
    }
}

// ---------------------------------------------------------------------------
// 1) GSE depthwise conv along W (k=3, pad 1).  The (0,3,1,2)->(0,2,3,1)
//    permutes cancel, so f shares x's layout; width neighbors are +-BANDS.
// ---------------------------------------------------------------------------
__global__ void gse_kernel(const float* __restrict__ x,
                           const float* __restrict__ wg,
                           const float* __restrict__ bg,
                           float* __restrict__ f)
{
    const int idx = blockIdx.x * blockDim.x + threadIdx.x;   // exact grid
    const int c = idx % BANDS;
    const int w = (idx / BANDS) % SPATIAL;
    const float w0 = wg[c * 3 + 0], w1 = wg[c * 3 + 1], w2 = wg[c * 3 + 2];
    float acc = bg[c] + x[idx] * w1;
    if (w > 0)           acc += x[idx - BANDS] * w0;
    if (w < SPATIAL - 1) acc += x[idx + BANDS] * w2;
    f[idx] = acc;
}

// ---------------------------------------------------------------------------
// 2) LayerNorm, one block per batch row, biased variance (E[x^2]-mu^2).
// ---------------------------------------------------------------------------
__global__ void ln_kernel(const float* __restrict__ f,
                          const float* __restrict__ g,
                          const float* __restrict__ beta,
                          float* __restrict__ out)
{
    __shared__ float s1[256], s2[256];
    const int row = blockIdx.x;
    const int tid = threadIdx.x;
    const float* fr = f + (size_t)row * DIN;
    float sum = 0.f, sq = 0.f;
    for (int i = tid; i < DIN; i += 256) { float v = fr[i]; sum += v; sq += v * v; }
    s1[tid] = sum; s2[tid] = sq;
    __syncthreads();
    for (int s = 128; s > 0; s >>= 1) {
        if (tid < s) { s1[tid] += s1[tid + s]; s2[tid] += s2[tid + s]; }
        __syncthreads();
    }
    const float mu  = s1[0] * (1.f / DIN);
    const float var = s2[0] * (1.f / DIN) - mu * mu;
    const float rs  = rsqrtf(var + LN_EPS);
    float* orow = out + (size_t)row * DIN;
    for (int i = tid; i < DIN; i += 256)
        orow[i] = (fr[i] - mu) * rs * g[i] + beta[i];
}

// ---------------------------------------------------------------------------
// 3) Split-K projection: xh = ln@Wx^T, zh = ln@Wz^T.  2 mats x 64 N-tiles x
//    10 uniform 1620-wide K-chunks = 1280 waves (320 blocks): enough in-flight
//    loads to run the 133 MB weight stream at HBM/L2 rate.  Constant chunk ->
//    fully pipelined inner loop, no tail, no EXEC-masked looping.
// ---------------------------------------------------------------------------
__global__ void proj_splitk_kernel(const float* __restrict__ ln,
                                   const float* __restrict__ Wx,
                                   const float* __restrict__ Wz,
                                   float* __restrict__ part)  // [2][10][32][1024]
{
    const int gw  = (blockIdx.x * blockDim.x + threadIdx.x) >> 5;
    const int mat = gw / (64 * PROJ_SPLIT);          // 0: Wx, 1: Wz
    const int r   = gw % (64 * PROJ_SPLIT);
    const int nt  = r / PROJ_SPLIT;
    const int ks  = r % PROJ_SPLIT;
    const int n0  = nt * 16;
    const int kb  = ks * PROJ_KCHUNK;
    const float* W = mat ? Wz : Wx;
    v8f c0 = {}, c1 = {};
    gemm32_tile<PROJ_KCHUNK>(ln + kb, DIN, W, DIN, 1, kb, n0, c0, c1);
    store_tile(part + (size_t)(mat * PROJ_SPLIT + ks) * BATCH * HIDDEN, HIDDEN,
               n0, c0, c1);
}

// ---------------------------------------------------------------------------
// 4) Split-K center-tap conv GEMMs: cf = xh@Wcf[:,:,1]^T, cb = zh@Wcb[:,:,1]^T.
//    Constant 128-wide K chunks, 1024 waves.
// ---------------------------------------------------------------------------
__global__ void conv_splitk_kernel(const float* __restrict__ xh,
                                   const float* __restrict__ zh,
                                   const float* __restrict__ Wcf,
                                   const float* __restrict__ Wcb,
                                   float* __restrict__ part)  // [2][8][32][1024]
{
    const int gw  = (blockIdx.x * blockDim.x + threadIdx.x) >> 5;
    const int mat = gw >> 9;
    const int nt  = (gw >> 3) & 63;
    const int ks  = gw & 7;
    const int n0  = nt * 16;
    const int kb  = ks * CONV_KCHUNK;
    const float* Act = mat ? zh : xh;
    const float* W   = mat ? Wcb : Wcf;
    v8f c0 = {}, c1 = {};
    gemm32_tile<CONV_KCHUNK>(Act + kb, HIDDEN, W, 3 * HIDDEN, 3,
                             (long)kb * 3 + 1, n0, c0, c1);
    store_tile(part + (size_t)(mat * CONV_SPLIT + ks) * BATCH * HIDDEN, HIDDEN,
               n0, c0, c1);
}

// ---------------------------------------------------------------------------
// Deterministic split-K reduction: out[b][n] = sum_s part[s][b][n] + bias(+bias2).
// ---------------------------------------------------------------------------
__global__ void reduce_kernel(const float* __restrict__ part, int nparts, int width,
                              const float* __restrict__ bias,
                              const float* __restrict__ bias2,
                              float* __restrict__ out)
{
    const int idx = blockIdx.x * blockDim.x + threadIdx.x;
    if (idx >= BATCH * width) return;
    const int n = idx % width;
    float s = 0.f;
    for (int p = 0; p < nparts; ++p) s += part[(size_t)p * BATCH * width + idx];
    s += bias[n];
    if (bias2) s += bias2[n];
    out[idx] = s;
}

// ---------------------------------------------------------------------------
// 5) fr[b,i] = mean_j tanh(cf[b,i] + A[i,j]*delta[i]).  One wave per (mat, i);
//    each lane streams its slice of row A[i] once, reused across all 32 batch
//    rows (A/Bm read exactly once), 67M native v_tanh_f32 total.
// ---------------------------------------------------------------------------
__global__ void tanh_mean_kernel(const float* __restrict__ cf,
                                 const float* __restrict__ cb,
                                 const float* __restrict__ A,
                                 const float* __restrict__ Bm,
                                 const float* __restrict__ delta,
                                 float* __restrict__ fr,
                                 float* __restrict__ br)
{
    const int gw   = (blockIdx.x * blockDim.x + threadIdx.x) >> 5;
    const int lane = threadIdx.x & 31;
    const int mat  = gw >> 10;               // 0: (A, cf)->fr  1: (Bm, cb)->br
    const int i    = gw & (HIDDEN - 1);
    const float* M = mat ? Bm : A;
    const float* C = mat ? cb : cf;
    float* O       = mat ? br : fr;
    const float d  = delta[i];

    float cbuf[BATCH];
#pragma unroll
    for (int b = 0; b < BATCH; ++b) cbuf[b] = C[(size_t)b * HIDDEN + i];

    float acc[BATCH];
#pragma unroll
    for (int b = 0; b < BATCH; ++b) acc[b] = 0.f;

    const float* Mi = M + (size_t)i * HIDDEN;
    for (int j = lane; j < HIDDEN; j += 32) {
        const float ad = Mi[j] * d;
#pragma unroll
        for (int b = 0; b < BATCH; ++b) acc[b] += fast_tanh(cbuf[b] + ad);
    }

    float mine = 0.f;
#pragma unroll
    for (int b = 0; b < BATCH; ++b) {
        float v = acc[b];
        for (int offs = 16; offs > 0; offs >>= 1) v += __shfl_xor(v, offs, 32);
        if (lane == b) mine = v;             // lane b keeps batch-row b's total
    }
    O[(size_t)lane * HIDDEN + i] = mine * (1.0f / HIDDEN);
}

// ---------------------------------------------------------------------------
// 6) Split-K output GEMM: y_part = fr@Wf^T + br@Wb^T, both chunks accumulated
//    into the same WMMA C registers.  32 N-tiles x 8 K-chunks = 256 waves.
// ---------------------------------------------------------------------------
__global__ void out_splitk_kernel(const float* __restrict__ fr,
                                  const float* __restrict__ br,
                                  const float* __restrict__ Wf,
                                  const float* __restrict__ Wb,
                                  float* __restrict__ part)  // [8][32][512]
{
    const int gw = (blockIdx.x * blockDim.x + threadIdx.x) >> 5;
    const int nt = gw >> 3;                  // 0..31
    const int ks = gw & 7;
    const int n0 = nt * 16;
    const int kb = ks * CONV_KCHUNK;
    v8f c0 = {}, c1 = {};
    gemm32_tile<CONV_KCHUNK>(fr + kb, HIDDEN, Wf, HIDDEN, 1, kb, n0, c0, c1);
    gemm32_tile<CONV_KCHUNK>(br + kb, HIDDEN, Wb, HIDDEN, 1, kb, n0, c0, c1);
    store_tile(part + (size_t)ks * BATCH * OUT_DIM, OUT_DIM, n0, c0, c1);
}

// ---------------------------------------------------------------------------
extern "C" void kernel_launch(void* const* d_in, const int* in_sizes, int n_in,
                              void* d_out, int out_size, void* d_ws, size_t ws_size,
                              hipStream_t stream)
{
    const float* x     = (const float*)d_in[0];
    const float* wgse  = (const float*)d_in[1];
    const float* bgse  = (const float*)d_in[2];
    const float* lng   = (const float*)d_in[3];
    const float* lnb   = (const float*)d_in[4];
    const float* Wx    = (const float*)d_in[5];
    const float* bx    = (const float*)d_in[6];
    const float* Wz    = (const float*)d_in[7];
    const float* bz    = (const float*)d_in[8];
    const float* Wcf   = (const float*)d_in[9];
    const float* bcf   = (const float*)d_in[10];
    const float* Wcb   = (const float*)d_in[11];
    const float* bcb   = (const float*)d_in[12];
    const float* A     = (const float*)d_in[13];
    const float* Bm    = (const float*)d_in[14];
    const float* delta = (const float*)d_in[15];
    const float* Wf    = (const float*)d_in[16];
    const float* bf    = (const float*)d_in[17];
    const float* Wb    = (const float*)d_in[18];
    const float* bb    = (const float*)d_in[19];
    float* y = (float*)d_out;

    float* f    = (float*)d_ws;                     // [32][16200]
    float* ln   = f    + (size_t)BATCH * DIN;       // [32][16200]
    float* xh   = ln   + (size_t)BATCH * DIN;       // [32][1024]
    float* zh   = xh   + BATCH * HIDDEN;
    float* cf   = zh   + BATCH * HIDDEN;
    float* cb   = cf   + BATCH * HIDDEN;
    float* fr   = cb   + BATCH * HIDDEN;
    float* br   = fr   + BATCH * HIDDEN;
    float* part = br   + BATCH * HIDDEN;            // [2][10][32][1024] = 655360 f

    gse_kernel<<<(BATCH * DIN) / 256, 256, 0, stream>>>(x, wgse, bgse, f);   // 2025 blocks
    ln_kernel<<<BATCH, 256, 0, stream>>>(f, lng, lnb, ln);

    proj_splitk_kernel<<<320, 128, 0, stream>>>(ln, Wx, Wz, part);           // 1280 waves
    reduce_kernel<<<128, 256, 0, stream>>>(part, PROJ_SPLIT, HIDDEN, bx, nullptr, xh);
    reduce_kernel<<<128, 256, 0, stream>>>(part + (size_t)PROJ_SPLIT * BATCH * HIDDEN,
                                           PROJ_SPLIT, HIDDEN, bz, nullptr, zh);

    conv_splitk_kernel<<<256, 128, 0, stream>>>(xh, zh, Wcf, Wcb, part);     // 1024 waves
    reduce_kernel<<<128, 256, 0, stream>>>(part, CONV_SPLIT, HIDDEN, bcf, nullptr, cf);
    reduce_kernel<<<128, 256, 0, stream>>>(part + (size_t)CONV_SPLIT * BATCH * HIDDEN,
                                           CONV_SPLIT, HIDDEN, bcb, nullptr, cb);

    tanh_mean_kernel<<<256, 256, 0, stream>>>(cf, cb, A, Bm, delta, fr, br); // 2048 waves

    out_splitk_kernel<<<64, 128, 0, stream>>>(fr, br, Wf, Wb, part);         // 256 waves
    reduce_kernel<<<64, 256, 0, stream>>>(part, CONV_SPLIT, OUT_DIM, bf, bb, y);
}